// HANLayer_8332236554727
// MI455X (gfx1250) — compile-verified
//
#include <hip/hip_runtime.h>
#include <math.h>

#define N_NODES 50000
#define P_PATHS 3
#define H_HEADS 4
#define F_DIM   128   // H*D
#define K_DIM   128   // input feature dim

typedef __attribute__((ext_vector_type(2))) float v2f;
typedef __attribute__((ext_vector_type(8))) float v8f;

// ---------------- init: zero accumulators, -inf for segment max ----------------
__global__ void k_init(float* __restrict__ zbuf, float* __restrict__ mmax,
                       float* __restrict__ denom, float* __restrict__ wsum) {
  size_t i = (size_t)blockIdx.x * blockDim.x + threadIdx.x;
  size_t totZ = (size_t)P_PATHS * N_NODES * F_DIM;
  if (i < totZ) zbuf[i] = 0.0f;
  size_t totH = (size_t)P_PATHS * N_NODES * H_HEADS;
  if (i < totH) { mmax[i] = -3.402823466e38f; denom[i] = 0.0f; }
  if (i < 4) wsum[i] = 0.0f;
}

// ---------------- feat[p] = h @ W[p]  via V_WMMA_F32_16X16X4_F32 ----------------
// one wave computes a 16-row x 128-col tile (8 accumulator tiles, K=128 in steps of 4)
__global__ void k_feat_gemm(const float* __restrict__ h,
                            const float* __restrict__ W,
                            float* __restrict__ feat) {
  const int wave = threadIdx.x >> 5;
  const int lane = threadIdx.x & 31;
  const int p    = blockIdx.y;
  const int tile = blockIdx.x * (blockDim.x >> 5) + wave;
  const int numTiles = N_NODES / 16;           // 3125 (exact)
  if (tile >= numTiles) return;                // wave-uniform: EXEC stays all-ones
  const int m0   = tile * 16;
  const int half = lane >> 4;                  // 0: K={0,1}, 1: K={2,3}
  const int lidx = lane & 15;                  // A row / B column within tile
  const int koff = half << 1;

  const float* __restrict__ Wp   = W + (size_t)p * K_DIM * F_DIM;
  const float* __restrict__ Arow = h + (size_t)(m0 + lidx) * K_DIM;

  v8f acc[8] = {};
  for (int k0 = 0; k0 < K_DIM; k0 += 4) {
    v2f a = *(const v2f*)(Arow + k0 + koff);   // contiguous K pair, 8B load
    const float* B0 = Wp + (size_t)(k0 + koff)     * F_DIM;
    const float* B1 = Wp + (size_t)(k0 + koff + 1) * F_DIM;
#pragma unroll
    for (int nt = 0; nt < 8; ++nt) {
      v2f b; b.x = B0[nt * 16 + lidx]; b.y = B1[nt * 16 + lidx];
      acc[nt] = __builtin_amdgcn_wmma_f32_16x16x4_f32(
          false, a, false, b, (short)0, acc[nt], false, false);
    }
  }
  float* __restrict__ out = feat + ((size_t)p * N_NODES + m0) * F_DIM;
#pragma unroll
  for (int nt = 0; nt < 8; ++nt)
#pragma unroll
    for (int v = 0; v < 8; ++v) {
      int row = v + (half << 3);
      out[(size_t)row * F_DIM + nt * 16 + lidx] = acc[nt][v];
    }
}

// ---------------- el/er: per-node per-head dot with attn vectors ----------------
// one 128-thread block per (p,n); each 32-lane wave reduces one head (D=32)
__global__ void k_el_er(const float* __restrict__ feat,
                        const float* __restrict__ attn_l,
                        const float* __restrict__ attn_r,
                        float* __restrict__ el, float* __restrict__ er) {
  int bid = blockIdx.x;                        // 0 .. P*N-1
  int p = bid / N_NODES;
  int n = bid - p * N_NODES;
  int t = threadIdx.x;                         // 0..127, head = t>>5
  size_t nb = (size_t)p * N_NODES + n;
  float f  = feat[nb * F_DIM + t];
  float sl = f * attn_l[p * F_DIM + t];
  float sr = f * attn_r[p * F_DIM + t];
  for (int m = 16; m >= 1; m >>= 1) {
    sl += __shfl_xor(sl, m, 32);
    sr += __shfl_xor(sr, m, 32);
  }
  if ((t & 31) == 0) {
    el[nb * H_HEADS + (t >> 5)] = sl;
    er[nb * H_HEADS + (t >> 5)] = sr;
  }
}

__device__ __forceinline__ float leaky(float x) { return x > 0.0f ? x : 0.2f * x; }

// ---------------- pass 1: segment max over dst ----------------
__global__ void k_edge_max(const int* __restrict__ src, const int* __restrict__ dst,
                           const float* __restrict__ el, const float* __restrict__ er,
                           float* __restrict__ mmax, int E) {
  long long i = (long long)blockIdx.x * blockDim.x + threadIdx.x;
  if (i >= (long long)P_PATHS * E) return;
  int p = (int)(i / E);
  long long e = i - (long long)p * E;
  int s = src[(size_t)p * E + e];
  int d = dst[(size_t)p * E + e];
  const float* elp = el + ((size_t)p * N_NODES + s) * H_HEADS;
  const float* erp = er + ((size_t)p * N_NODES + d) * H_HEADS;
  float* mp = mmax + ((size_t)p * N_NODES + d) * H_HEADS;
#pragma unroll
  for (int hh = 0; hh < H_HEADS; ++hh)
    atomicMax(&mp[hh], leaky(elp[hh] + erp[hh]));
}

// ---------------- pass 2: segment sum of exp(e - m) ----------------
__global__ void k_edge_sum(const int* __restrict__ src, const int* __restrict__ dst,
                           const float* __restrict__ el, const float* __restrict__ er,
                           const float* __restrict__ mmax, float* __restrict__ denom, int E) {
  long long i = (long long)blockIdx.x * blockDim.x + threadIdx.x;
  if (i >= (long long)P_PATHS * E) return;
  int p = (int)(i / E);
  long long e = i - (long long)p * E;
  int s = src[(size_t)p * E + e];
  int d = dst[(size_t)p * E + e];
  const float* elp = el + ((size_t)p * N_NODES + s) * H_HEADS;
  const float* erp = er + ((size_t)p * N_NODES + d) * H_HEADS;
  const float* mp  = mmax + ((size_t)p * N_NODES + d) * H_HEADS;
  float* dp = denom + ((size_t)p * N_NODES + d) * H_HEADS;
#pragma unroll
  for (int hh = 0; hh < H_HEADS; ++hh)
    atomicAdd(&dp[hh], expf(leaky(elp[hh] + erp[hh]) - mp[hh]));
}

// ---------------- pass 3: weighted aggregation; 128 threads per edge ----------------
__global__ void k_edge_aggr(const int* __restrict__ src, const int* __restrict__ dst,
                            const float* __restrict__ el, const float* __restrict__ er,
                            const float* __restrict__ mmax, const float* __restrict__ denom,
                            const float* __restrict__ feat, float* __restrict__ zbuf, int E) {
  int p = blockIdx.y;
  long long e = (long long)blockIdx.x * (blockDim.x / F_DIM) + (threadIdx.x / F_DIM);
  if (e >= E) return;
  int t  = threadIdx.x & (F_DIM - 1);
  int hh = t >> 5;
  int s = src[(size_t)p * E + e];
  int d = dst[(size_t)p * E + e];
  size_t nbs = (size_t)p * N_NODES + s;
  size_t nbd = (size_t)p * N_NODES + d;
  float ev = leaky(el[nbs * H_HEADS + hh] + er[nbd * H_HEADS + hh]);
  float alpha = expf(ev - mmax[nbd * H_HEADS + hh]) / denom[nbd * H_HEADS + hh];
  atomicAdd(&zbuf[nbd * F_DIM + t], feat[nbs * F_DIM + t] * alpha);
}

// ---------------- bias + ELU (in place on z) ----------------
__global__ void k_bias_elu(float* __restrict__ zbuf, const float* __restrict__ bias) {
  size_t i = (size_t)blockIdx.x * blockDim.x + threadIdx.x;
  size_t tot = (size_t)P_PATHS * N_NODES * F_DIM;
  if (i >= tot) return;
  int p   = (int)(i / ((size_t)N_NODES * F_DIM));
  int col = (int)(i & (F_DIM - 1));
  float x = zbuf[i] + bias[p * F_DIM + col];
  zbuf[i] = x > 0.0f ? x : expm1f(x);
}

// ---------------- semantic score: wsum[p] = mean_n( tanh(z@W1+b1)@W2 ) ----------------
// WMMA f32 16-row tile; fused tanh*W2 + wave reduction + single atomicAdd per wave
__global__ void k_semantic(const float* __restrict__ zbuf,
                           const float* __restrict__ W1,
                           const float* __restrict__ b1,
                           const float* __restrict__ W2,
                           float* __restrict__ wsum) {
  const int wave = threadIdx.x >> 5;
  const int lane = threadIdx.x & 31;
  const int tile = blockIdx.x * (blockDim.x >> 5) + wave;
  const int numTiles = (N_NODES * P_PATHS) / 16;   // 9375 (exact; tile never spans p)
  if (tile >= numTiles) return;
  const int r0   = tile * 16;
  const int p    = r0 / N_NODES;
  const int half = lane >> 4;
  const int lidx = lane & 15;
  const int koff = half << 1;
  const float* __restrict__ Arow = zbuf + (size_t)(r0 + lidx) * F_DIM;

  v8f acc[8] = {};
  for (int k0 = 0; k0 < F_DIM; k0 += 4) {
    v2f a = *(const v2f*)(Arow + k0 + koff);
    const float* B0 = W1 + (size_t)(k0 + koff)     * F_DIM;
    const float* B1 = W1 + (size_t)(k0 + koff + 1) * F_DIM;
#pragma unroll
    for (int nt = 0; nt < 8; ++nt) {
      v2f b; b.x = B0[nt * 16 + lidx]; b.y = B1[nt * 16 + lidx];
      acc[nt] = __builtin_amdgcn_wmma_f32_16x16x4_f32(
          false, a, false, b, (short)0, acc[nt], false, false);
    }
  }
  float partial = 0.0f;
#pragma unroll
  for (int nt = 0; nt < 8; ++nt) {
    int col = nt * 16 + lidx;
    float w2 = W2[col];
    float bb = b1[col];
#pragma unroll
    for (int v = 0; v < 8; ++v)
      partial += tanhf(acc[nt][v] + bb) * w2;
  }
  for (int m = 16; m >= 1; m >>= 1) partial += __shfl_xor(partial, m, 32);
  if (lane == 0) atomicAdd(&wsum[p], partial * (1.0f / N_NODES));
}

// ---------------- softmax over P=3 ----------------
__global__ void k_beta(const float* __restrict__ wsum, float* __restrict__ beta) {
  if (threadIdx.x == 0 && blockIdx.x == 0) {
    float w0 = wsum[0], w1 = wsum[1], w2 = wsum[2];
    float mx = fmaxf(w0, fmaxf(w1, w2));
    float e0 = expf(w0 - mx), e1 = expf(w1 - mx), e2 = expf(w2 - mx);
    float s = e0 + e1 + e2;
    beta[0] = e0 / s; beta[1] = e1 / s; beta[2] = e2 / s;
  }
}

// ---------------- final blend ----------------
__global__ void k_combine(const float* __restrict__ zbuf, const float* __restrict__ beta,
                          float* __restrict__ out) {
  size_t i = (size_t)blockIdx.x * blockDim.x + threadIdx.x;
  size_t tot = (size_t)N_NODES * F_DIM;
  if (i >= tot) return;
  size_t stride = (size_t)N_NODES * F_DIM;
  out[i] = beta[0] * zbuf[i] + beta[1] * zbuf[i + stride] + beta[2] * zbuf[i + 2 * stride];
}

extern "C" void kernel_launch(void* const* d_in, const int* in_sizes, int n_in,
                              void* d_out, int out_size, void* d_ws, size_t ws_size,
                              hipStream_t stream) {
  const float* h      = (const float*)d_in[0];
  const float* W      = (const float*)d_in[1];
  const float* attn_l = (const float*)d_in[2];
  const float* attn_r = (const float*)d_in[3];
  const float* bias   = (const float*)d_in[4];
  const float* W1     = (const float*)d_in[5];
  const float* b1     = (const float*)d_in[6];
  const float* W2     = (const float*)d_in[7];
  const int*   src    = (const int*)d_in[8];
  const int*   dst    = (const int*)d_in[9];
  const int E = in_sizes[8] / P_PATHS;   // 850000

  // workspace layout (floats): feat | z | el | er | mmax | denom | wsum[4] | beta[4]
  float* ws    = (float*)d_ws;
  size_t featN = (size_t)P_PATHS * N_NODES * F_DIM;
  size_t hN    = (size_t)P_PATHS * N_NODES * H_HEADS;
  float* feat  = ws;
  float* zbuf  = feat  + featN;
  float* el    = zbuf  + featN;
  float* er    = el    + hN;
  float* mmax  = er    + hN;
  float* denom = mmax  + hN;
  float* wsum  = denom + hN;
  float* beta  = wsum  + 4;

  { // init (covers largest array; re-run every launch for graph replay)
    int blocks = (int)((featN + 255) / 256);
    k_init<<<blocks, 256, 0, stream>>>(zbuf, mmax, denom, wsum);
  }
  { // feat = h @ W[p]  (WMMA f32)
    dim3 grid((N_NODES / 16 + 7) / 8, P_PATHS);
    k_feat_gemm<<<grid, 256, 0, stream>>>(h, W, feat);
  }
  k_el_er<<<P_PATHS * N_NODES, 128, 0, stream>>>(feat, attn_l, attn_r, el, er);
  {
    long long tot = (long long)P_PATHS * E;
    int blocks = (int)((tot + 255) / 256);
    k_edge_max<<<blocks, 256, 0, stream>>>(src, dst, el, er, mmax, E);
    k_edge_sum<<<blocks, 256, 0, stream>>>(src, dst, el, er, mmax, denom, E);
  }
  { // 2 edges per 256-thread block, coalesced 128-float gather + scatter-add
    dim3 grid((E + 1) / 2, P_PATHS);
    k_edge_aggr<<<grid, 256, 0, stream>>>(src, dst, el, er, mmax, denom, feat, zbuf, E);
  }
  {
    int blocks = (int)((featN + 255) / 256);
    k_bias_elu<<<blocks, 256, 0, stream>>>(zbuf, bias);
  }
  { // semantic attention scores (WMMA f32, fused reduction)
    int tiles = (N_NODES * P_PATHS) / 16;
    k_semantic<<<(tiles + 7) / 8, 256, 0, stream>>>(zbuf, W1, b1, W2, wsum);
  }
  k_beta<<<1, 32, 0, stream>>>(wsum, beta);
  {
    size_t tot = (size_t)N_NODES * F_DIM;
    k_combine<<<(int)((tot + 255) / 256), 256, 0, stream>>>(zbuf, beta, (float*)d_out);
  }
}